// MixtureOfGaussians_47356309406121
// MI455X (gfx1250) — compile-verified
//
#include <hip/hip_runtime.h>
#include <hip/hip_bf16.h>

// MoG log-likelihood via fp32 WMMA (V_WMMA_F32_16X16X4_F32) on gfx1250.
// B=4096 points, D=64 dims, K=1000 components.
// (z-m)^2/v expanded so the B*K*D term is a 4096x1008 GEMM, inner dim 128.
// GEMM kernel register-blocks 4 row-tiles per wave: B fragment reused 4x,
// 128 WMMAs per wave, 1.25 loads/WMMA.

typedef __attribute__((ext_vector_type(2))) float v2f;
typedef __attribute__((ext_vector_type(8))) float v8f;

#define NB    4096
#define ND    64
#define NK    1000
#define NKP   1008          // padded to 63 tiles of 16
#define KDIM  128           // inner dim: [z^2 | z]
#define NCT   63            // col tiles
#define MBLK  4             // row tiles per wave (64 rows)

// ---------------- kernel 1a: pack A = [z^2 | z], row-major 4096x128 ----------
__global__ void __launch_bounds__(256) pack_a_kernel(const float* __restrict__ z,
                                                     float* __restrict__ Apack) {
  int idx = blockIdx.x * 256 + threadIdx.x;     // 0 .. 4096*64-1
  int b = idx >> 6;
  int d = idx & 63;
  float zv = z[idx];
  Apack[b * KDIM + d]      = zv * zv;
  Apack[b * KDIM + 64 + d] = zv;
}

// ---------------- kernel 1b: pack B (transposed, pair-interleaved) + bias ----
// B matrix (128 x NKP): rows 0..63 = w[k][d], rows 64..127 = -2*m*w.
// Storage: float2 Bpack[p][c] = {row 2p, row 2p+1} of column c,  p in [0,64).
// bias[k] = -32*ln(2pi) - 0.5*sum(log v) - 0.5*sum(m^2 w);  bias[pad] = -inf.
__global__ void __launch_bounds__(64) pack_b_kernel(const float* __restrict__ z_pre,
                                                    float* __restrict__ Bpack,
                                                    float* __restrict__ bias) {
  int k = blockIdx.x;       // 0 .. 1007
  int d = threadIdx.x;      // 0 .. 63

  float wv  = 0.f;   // row d value
  float mwv = 0.f;   // row 64+d value
  float lv  = 0.f;   // log(v)
  float mq  = 0.f;   // m^2 * w
  if (k < NK) {
    float m = z_pre[k * ND + d];
    float h = z_pre[(NK + k) * ND + d];
    float sp = fmaxf(h, 0.f) + log1pf(expf(-fabsf(h)));   // softplus, stable
    float v  = sp + 1e-8f;
    wv  = 1.0f / v;
    mwv = -2.0f * m * wv;
    lv  = logf(v);
    mq  = m * m * wv;
  }
  // row d
  Bpack[((d >> 1) * NKP + k) * 2 + (d & 1)] = wv;
  // row 64+d
  int d2 = 64 + d;
  Bpack[((d2 >> 1) * NKP + k) * 2 + (d2 & 1)] = mwv;

  __shared__ float slog[64];
  __shared__ float smq[64];
  slog[d] = lv;
  smq[d]  = mq;
  __syncthreads();
  for (int s = 32; s > 0; s >>= 1) {
    if (d < s) { slog[d] += slog[d + s]; smq[d] += smq[d + s]; }
    __syncthreads();
  }
  if (d == 0) {
    if (k < NK)
      bias[k] = -58.81206613f /* -32*ln(2*pi) */ - 0.5f * slog[0] - 0.5f * smq[0];
    else
      bias[k] = -__builtin_inff();
  }
}

// ---------------- kernel 2: WMMA GEMM + per-tile online logsumexp ------------
// One wave per 64x16 output block (4 row-tiles x 1 col-tile).
// Per k-step: 1 B-fragment load shared across 4 WMMAs, 4 A-fragment loads.
// C layout: VGPR i <-> row (i + 8*(lane/16)), lane%16 <-> col.
// A frag:   VGPR j <-> K = 4*kk + 2*(lane/16) + j, lane%16 <-> row.
// B frag:   VGPR j <-> K = 4*kk + 2*(lane/16) + j, lane%16 <-> col.
__global__ void __launch_bounds__(256) gemm_lse_partial_kernel(
    const float* __restrict__ Apack, const float* __restrict__ Bpack,
    const float* __restrict__ bias,
    float* __restrict__ pmax, float* __restrict__ psum) {
  const int lane  = threadIdx.x & 31;
  const int wid   = blockIdx.x * 8 + (threadIdx.x >> 5);
  const int ct    = wid % NCT;             // col tile
  const int rb    = wid / NCT;             // row block (0..63), 64 rows each
  const int g     = lane >> 4;             // half-wave group
  const int l16   = lane & 15;
  const int row0  = rb * (16 * MBLK) + l16; // first A row this lane loads
  const int col   = ct * 16 + l16;          // component column this lane owns

  const float* aptr = Apack + row0 * KDIM + 2 * g;   // +4 floats per kk, +16*KDIM per tile
  const float* bptr = Bpack + (g * NKP + col) * 2;   // +4032 floats per kk

  v8f c0 = {}, c1 = {}, c2 = {}, c3 = {};
#pragma unroll
  for (int kk = 0; kk < 32; ++kk) {
    v2f b  = *(const v2f*)(bptr + kk * (2 * NKP * 2));
    v2f a0 = *(const v2f*)(aptr + kk * 4);
    v2f a1 = *(const v2f*)(aptr + kk * 4 + 16 * KDIM);
    v2f a2 = *(const v2f*)(aptr + kk * 4 + 32 * KDIM);
    v2f a3 = *(const v2f*)(aptr + kk * 4 + 48 * KDIM);
    c0 = __builtin_amdgcn_wmma_f32_16x16x4_f32(false, a0, false, b, (short)0, c0, false, false);
    c1 = __builtin_amdgcn_wmma_f32_16x16x4_f32(false, a1, false, b, (short)0, c1, false, false);
    c2 = __builtin_amdgcn_wmma_f32_16x16x4_f32(false, a2, false, b, (short)0, c2, false, false);
    c3 = __builtin_amdgcn_wmma_f32_16x16x4_f32(false, a3, false, b, (short)0, c3, false, false);
  }

  const float bv = bias[col];              // -inf for padded columns
  v8f acc[4] = {c0, c1, c2, c3};
#pragma unroll
  for (int t = 0; t < MBLK; ++t) {
#pragma unroll
    for (int i = 0; i < 8; ++i) {
      float x = bv - 0.5f * acc[t][i];     // log_prob (row rb*64+t*16+i+8g, col)
      // row max across the 16 lanes of this half-wave
      float m = x;
      m = fmaxf(m, __shfl_xor(m, 1, 32));
      m = fmaxf(m, __shfl_xor(m, 2, 32));
      m = fmaxf(m, __shfl_xor(m, 4, 32));
      m = fmaxf(m, __shfl_xor(m, 8, 32));
      float e = expf(x - m);               // 0 for padded cols (x = -inf)
      e += __shfl_xor(e, 1, 32);
      e += __shfl_xor(e, 2, 32);
      e += __shfl_xor(e, 4, 32);
      e += __shfl_xor(e, 8, 32);
      if (l16 == 0) {
        int orow = rb * (16 * MBLK) + t * 16 + i + 8 * g;
        pmax[orow * NCT + ct] = m;
        psum[orow * NCT + ct] = e;
      }
    }
  }
}

// ---------------- kernel 3: combine per-tile partials ------------------------
__global__ void __launch_bounds__(256) lse_combine_kernel(
    const float* __restrict__ pmax, const float* __restrict__ psum,
    float* __restrict__ out) {
  int b = blockIdx.x * 256 + threadIdx.x;   // 0..4095
  const float* pm = pmax + b * NCT;
  const float* ps = psum + b * NCT;
  float M = -__builtin_inff();
  for (int t = 0; t < NCT; ++t) M = fmaxf(M, pm[t]);
  float S = 0.f;
  for (int t = 0; t < NCT; ++t) S += ps[t] * expf(pm[t] - M);
  out[b] = M + logf(S) - 6.907755279f;      // - ln(1000)
}

// ---------------- launch -----------------------------------------------------
extern "C" void kernel_launch(void* const* d_in, const int* in_sizes, int n_in,
                              void* d_out, int out_size, void* d_ws, size_t ws_size,
                              hipStream_t stream) {
  const float* z     = (const float*)d_in[0];   // (4096, 64)
  const float* z_pre = (const float*)d_in[1];   // (1, 2000, 64)
  float* out = (float*)d_out;                   // (4096,)

  char* ws = (char*)d_ws;
  // offsets (bytes), all >= 8-aligned
  float* Apack = (float*)(ws + 0);                      // 4096*128*4   = 2,097,152
  float* Bpack = (float*)(ws + 2097152);                // 64*1008*2*4  =   516,096
  float* bias  = (float*)(ws + 2097152 + 516096);       // 1008*4 (pad to 4096)
  float* pmax  = (float*)(ws + 2617344);                // 4096*63*4    = 1,032,192
  float* psum  = (float*)(ws + 2617344 + 1032192);      // 4096*63*4

  pack_a_kernel<<<(NB * ND) / 256, 256, 0, stream>>>(z, Apack);
  pack_b_kernel<<<NKP, 64, 0, stream>>>(z_pre, Bpack, bias);
  // 64 row-blocks * 63 col-tiles = 4032 waves, 8 waves/block -> 504 blocks
  // (exact coverage: EXEC stays all-ones through every WMMA)
  gemm_lse_partial_kernel<<<504, 256, 0, stream>>>(Apack, Bpack, bias, pmax, psum);
  lse_combine_kernel<<<NB / 256, 256, 0, stream>>>(pmax, psum, out);
}